// GCNConvBlock1_10161892622613
// MI455X (gfx1250) — compile-verified
//
#include <hip/hip_runtime.h>
#include <hip/hip_bf16.h>
#include <stdint.h>

typedef __attribute__((ext_vector_type(2))) float v2f;
typedef __attribute__((ext_vector_type(8))) float v8f;

#define D_FEAT 64

// ---------------------------------------------------------------- zero fills
// n must be a multiple of 4 (N*64 always is)
__global__ void zero_f32x4_kernel(float4* __restrict__ p, long long n4) {
    long long i = (long long)blockIdx.x * blockDim.x + threadIdx.x;
    if (i < n4) p[i] = make_float4(0.f, 0.f, 0.f, 0.f);
}
__global__ void zero_u32_kernel(unsigned* __restrict__ p, long long n) {
    long long i = (long long)blockIdx.x * blockDim.x + threadIdx.x;
    if (i < n) p[i] = 0u;
}

// ---------------------------------------------------------------- degrees
// deg[c] = number of edges with destination c (self-loop +1 added later)
__global__ void degree_kernel(const int* __restrict__ edge_index,
                              unsigned* __restrict__ deg, int E) {
    int e = blockIdx.x * blockDim.x + threadIdx.x;
    if (e >= E) return;
    int c = edge_index[(long long)E + e];   // col = destination
    atomicAdd(&deg[c], 1u);
}

// dinv[n] = rsqrt(deg[n] + 1)   (+1 = self loop; always > 0)
__global__ void dinv_kernel(const unsigned* __restrict__ deg,
                            float* __restrict__ dinv, int N) {
    int n = blockIdx.x * blockDim.x + threadIdx.x;
    if (n >= N) return;
    dinv[n] = rsqrtf((float)(deg[n] + 1u));
}

// ---------------------------------------------------------------- WMMA GEMM
// xw[m, :] = x[m, :] @ W,  fp32 via V_WMMA_F32_16X16X4_F32.
// One wave computes a 16-row tile across all 64 output columns (4 N-tiles).
__global__ void gemm_wmma_kernel(const float* __restrict__ x,
                                 const float* __restrict__ W,
                                 float* __restrict__ xw, int ntiles) {
    int wave = (int)((blockIdx.x * blockDim.x + threadIdx.x) >> 5);
    int lane = threadIdx.x & 31;
    if (wave >= ntiles) return;                 // wave-uniform exit: EXEC stays all-ones

    const int m0   = wave * 16;
    const int mrow = lane & 15;                 // A row / D col index
    const int lhi  = lane >> 4;                 // lane-half select

    v8f acc0 = {}, acc1 = {}, acc2 = {}, acc3 = {};
    const float* xrow = x + (long long)(m0 + mrow) * D_FEAT;

#pragma unroll
    for (int k0 = 0; k0 < D_FEAT; k0 += 4) {
        // A fragment 16x4: lane holds x[mrow, k0 + 2*lhi + {0,1}]
        v2f a;
        a.x = xrow[k0 + 2 * lhi + 0];
        a.y = xrow[k0 + 2 * lhi + 1];
        // B fragments 4x16 per N-tile: VGPR v -> K = k0 + v + 2*lhi, col = lane&15
        v2f b0, b1, b2, b3;
        const float* Wk0 = W + (long long)(k0 + 2 * lhi + 0) * D_FEAT + mrow;
        const float* Wk1 = W + (long long)(k0 + 2 * lhi + 1) * D_FEAT + mrow;
        b0.x = Wk0[0];  b0.y = Wk1[0];
        b1.x = Wk0[16]; b1.y = Wk1[16];
        b2.x = Wk0[32]; b2.y = Wk1[32];
        b3.x = Wk0[48]; b3.y = Wk1[48];

        acc0 = __builtin_amdgcn_wmma_f32_16x16x4_f32(false, a, false, b0, (short)0, acc0, false, false);
        acc1 = __builtin_amdgcn_wmma_f32_16x16x4_f32(false, a, false, b1, (short)0, acc1, false, false);
        acc2 = __builtin_amdgcn_wmma_f32_16x16x4_f32(false, a, false, b2, (short)0, acc2, false, false);
        acc3 = __builtin_amdgcn_wmma_f32_16x16x4_f32(false, a, false, b3, (short)0, acc3, false, false);
    }

    // D layout: VGPR i, lane l -> row m0 + i + 8*lhi, col (l&15) + 16*ntile
#pragma unroll
    for (int i = 0; i < 8; ++i) {
        long long r = (long long)(m0 + i + 8 * lhi) * D_FEAT + mrow;
        xw[r +  0] = acc0[i];
        xw[r + 16] = acc1[i];
        xw[r + 32] = acc2[i];
        xw[r + 48] = acc3[i];
    }
}

// scalar tail for rows not covered by 16-row tiles (none for N=100000)
__global__ void gemm_tail_kernel(const float* __restrict__ x,
                                 const float* __restrict__ W,
                                 float* __restrict__ xw, int n0, int N) {
    int t = blockIdx.x * blockDim.x + threadIdx.x;
    int n = n0 + (t >> 6);
    int d = t & 63;
    if (n >= N) return;
    float s = 0.0f;
    for (int k = 0; k < D_FEAT; ++k) s += x[(long long)n * D_FEAT + k] * W[k * D_FEAT + d];
    xw[(long long)n * D_FEAT + d] = s;
}

// ---------------------------------------------------------------- scatter
// 16 threads per edge; lane handles 4 consecutive features via float4 gather
// (global_load_b128 from L2-resident xw) + 4 f32 atomics.
// out[col, d] += dinv[row]*dinv[col] * xw[row, d]
__global__ void scatter_kernel(const int* __restrict__ edge_index,
                               const float* __restrict__ dinv,
                               const float* __restrict__ xw,
                               float* __restrict__ out, int E) {
    long long t = (long long)blockIdx.x * blockDim.x + threadIdx.x;
    int e  = (int)(t >> 4);
    int d4 = (int)(t & 15) << 2;
    if (e >= E) return;
    int r = edge_index[e];
    int c = edge_index[(long long)E + e];
    float nrm = dinv[r] * dinv[c];
    const float4 v = *(const float4*)(xw + (long long)r * D_FEAT + d4);
    float* o = out + (long long)c * D_FEAT + d4;
    atomicAdd(o + 0, nrm * v.x);
    atomicAdd(o + 1, nrm * v.y);
    atomicAdd(o + 2, nrm * v.z);
    atomicAdd(o + 3, nrm * v.w);
}

// ---------------------------------------------------------------- finalize
// out = relu(out + bias + dinv[n]^2 * xw[n, :])   (self-loop folded here)
// 16 threads per node, float4 per thread.
__global__ void finalize_kernel(float* __restrict__ out,
                                const float* __restrict__ xw,
                                const float* __restrict__ dinv,
                                const float* __restrict__ bias, int N) {
    long long t = (long long)blockIdx.x * blockDim.x + threadIdx.x;
    int n  = (int)(t >> 4);
    int d4 = (int)(t & 15) << 2;
    if (n >= N) return;
    float di2 = dinv[n] * dinv[n];
    long long idx = (long long)n * D_FEAT + d4;
    float4 acc = *(float4*)(out + idx);
    float4 xv  = *(const float4*)(xw + idx);
    float4 bv  = *(const float4*)(bias + d4);
    float4 res;
    res.x = fmaxf(acc.x + bv.x + di2 * xv.x, 0.0f);
    res.y = fmaxf(acc.y + bv.y + di2 * xv.y, 0.0f);
    res.z = fmaxf(acc.z + bv.z + di2 * xv.z, 0.0f);
    res.w = fmaxf(acc.w + bv.w + di2 * xv.w, 0.0f);
    *(float4*)(out + idx) = res;
}

// ---------------------------------------------------------------- launcher
extern "C" void kernel_launch(void* const* d_in, const int* in_sizes, int n_in,
                              void* d_out, int out_size, void* d_ws, size_t ws_size,
                              hipStream_t stream) {
    const float* x          = (const float*)d_in[0];
    const int*   edge_index = (const int*)d_in[1];   // JAX default: int64 request -> int32 arrays
    const float* W          = (const float*)d_in[2];
    const float* bias       = (const float*)d_in[3];
    float*       out        = (float*)d_out;

    const int N = in_sizes[0] / D_FEAT;   // 100000
    const int E = in_sizes[1] / 2;        // 1600000

    // workspace layout (256B aligned): xw[N*64] f32 | deg[N] u32 | dinv[N] f32
    char* ws = (char*)d_ws;
    size_t off = 0;
    float*    xw   = (float*)(ws + off);   off += (((size_t)N * D_FEAT * 4) + 255) & ~(size_t)255;
    unsigned* deg  = (unsigned*)(ws + off); off += (((size_t)N * 4) + 255) & ~(size_t)255;
    float*    dinv = (float*)(ws + off);

    const int B = 256;
    long long outElems = (long long)N * D_FEAT;

    // 1) zero accumulators (d_out is poisoned by harness; deg must start at 0)
    long long out4 = outElems >> 2;
    zero_f32x4_kernel<<<(unsigned)((out4 + B - 1) / B), B, 0, stream>>>((float4*)out, out4);
    zero_u32_kernel<<<(unsigned)((N + B - 1) / B), B, 0, stream>>>(deg, N);

    // 2) degrees over destinations, then deg_inv_sqrt (self-loop folded in)
    degree_kernel<<<(unsigned)((E + B - 1) / B), B, 0, stream>>>(edge_index, deg, E);
    dinv_kernel<<<(unsigned)((N + B - 1) / B), B, 0, stream>>>(deg, dinv, N);

    // 3) xw = x @ W via fp32 WMMA (one wave per 16-row tile)
    int ntiles = N / 16;
    if (ntiles > 0) {
        long long threads = (long long)ntiles * 32;
        gemm_wmma_kernel<<<(unsigned)((threads + B - 1) / B), B, 0, stream>>>(x, W, xw, ntiles);
    }
    int rem = N - ntiles * 16;
    if (rem > 0) {
        long long threads = (long long)rem * D_FEAT;
        gemm_tail_kernel<<<(unsigned)((threads + B - 1) / B), B, 0, stream>>>(x, W, xw, ntiles * 16, N);
    }

    // 4) normalized scatter-add over edges (float4 gather + L2-resident atomics)
    long long sthreads = (long long)E * 16;
    scatter_kernel<<<(unsigned)((sthreads + B - 1) / B), B, 0, stream>>>(edge_index, dinv, xw, out, E);

    // 5) self-loop term + bias + ReLU (float4)
    long long fthreads = (long long)N * 16;
    finalize_kernel<<<(unsigned)((fthreads + B - 1) / B), B, 0, stream>>>(out, xw, dinv, bias, N);
}